// DeformConv1D_5875515261457
// MI455X (gfx1250) — compile-verified
//
#include <hip/hip_runtime.h>

// Problem geometry (fixed by the reference's setup_inputs): b = 2048, c = 4096.
#define C_DIM 4096
#define KSZ   3
#define PADN  1
#define BLOCK 256

// LDS layout: [0..2] unused, [3] = x_pad[0] (zero), [4 .. C+3] = x row,
// [C+4] = x_pad[C+1] (zero). Row starts at byte offset 16 -> B128-aligned.
#define LDS_FRONT 4
#define LDS_ELEMS (LDS_FRONT + C_DIM + 4)

__device__ __forceinline__ float dc_elem(const float* s_xpad, int ci, int k,
                                         float off, float cpad_m1) {
    // p = p_0 + p_n + offset;  p_0 = ci+1 (1-based), p_n = k-1
    float p  = (float)(ci + k) + off;           // (ci+1) + (k-1) + off
    float fp = floorf(p);
    float qlf = fminf(fmaxf(fp,        0.0f), cpad_m1);
    float qrf = fminf(fmaxf(fp + 1.0f, 0.0f), cpad_m1);
    int   ql  = (int)qlf;
    int   qr  = (int)qrf;
    // detached out-of-range mask: p < PAD  ||  p > c_pad-1-PAD
    bool  oob = (p < (float)PADN) || (p > cpad_m1 - (float)PADN);
    float pm  = oob ? fp : p;
    pm = fminf(fmaxf(pm, 0.0f), cpad_m1);
    float gl = 1.0f + (qlf - pm);
    float gr = 1.0f - (qrf - pm);
    return gl * s_xpad[ql] + gr * s_xpad[qr];   // ds_load gathers
}

__global__ __launch_bounds__(BLOCK) void deform_conv1d_kernel(
    const float* __restrict__ x,        // (b, C)
    const float* __restrict__ offset,   // (b, KSZ, C)
    const float* __restrict__ conv_w,   // (KSZ)
    float* __restrict__ out)            // (b, C)
{
    __shared__ float s_x[LDS_ELEMS];

    const int b   = blockIdx.x;
    const int tid = threadIdx.x;

    const float* xrow = x + (size_t)b * C_DIM;

    // ---- async-stage the 16KB x row into LDS (bypasses VGPRs, ASYNCcnt) ----
    {
        unsigned lds_byte = (unsigned)((LDS_FRONT + tid * 4) * 4); // 16B/lane, aligned
        unsigned goff     = (unsigned)(tid * 16);
        const char* gbase = (const char*)xrow;                     // uniform -> SGPR pair
        #pragma unroll
        for (int i = 0; i < C_DIM / (BLOCK * 4); ++i) {            // 4 iterations
            asm volatile("global_load_async_to_lds_b128 %0, %1, %2"
                         :
                         : "v"(lds_byte), "v"(goff), "s"(gbase)
                         : "memory");
            lds_byte += BLOCK * 16u;
            goff     += BLOCK * 16u;
        }
    }

    if (tid == 0) {
        s_x[LDS_FRONT - 1]     = 0.0f;   // x_pad[0]
        s_x[LDS_FRONT + C_DIM] = 0.0f;   // x_pad[C+1]
    }

    // wait for this wave's async copies, then make row visible to the block
    asm volatile("s_wait_asynccnt 0x0" ::: "memory");
    __syncthreads();

    const float w0 = conv_w[0];
    const float w1 = conv_w[1];
    const float w2 = conv_w[2];

    const float* offrow = offset + (size_t)b * (KSZ * C_DIM);
    float*       outrow = out    + (size_t)b * C_DIM;
    const float* s_xpad = &s_x[LDS_FRONT - 1];          // s_xpad[j] == x_pad[j]
    const float  cpad_m1 = (float)(C_DIM + 2 * PADN - 1);

    // 4 consecutive channels per thread -> float4 offset loads / float4 stores
    #pragma unroll
    for (int j = 0; j < C_DIM / (BLOCK * 4); ++j) {     // 4 iterations
        const int ci0 = j * (BLOCK * 4) + tid * 4;
        float acc0 = 0.0f, acc1 = 0.0f, acc2 = 0.0f, acc3 = 0.0f;
        #pragma unroll
        for (int k = 0; k < KSZ; ++k) {
            const float4 o4 = *(const float4*)(offrow + (size_t)k * C_DIM + ci0);
            const float  wk = (k == 0) ? w0 : (k == 1) ? w1 : w2;
            acc0 = fmaf(dc_elem(s_xpad, ci0 + 0, k, o4.x, cpad_m1), wk, acc0);
            acc1 = fmaf(dc_elem(s_xpad, ci0 + 1, k, o4.y, cpad_m1), wk, acc1);
            acc2 = fmaf(dc_elem(s_xpad, ci0 + 2, k, o4.z, cpad_m1), wk, acc2);
            acc3 = fmaf(dc_elem(s_xpad, ci0 + 3, k, o4.w, cpad_m1), wk, acc3);
        }
        *(float4*)(outrow + ci0) = make_float4(acc0, acc1, acc2, acc3);
    }
}

extern "C" void kernel_launch(void* const* d_in, const int* in_sizes, int n_in,
                              void* d_out, int out_size, void* d_ws, size_t ws_size,
                              hipStream_t stream) {
    const float* x      = (const float*)d_in[0];   // (b, c, 1, 1) contiguous
    const float* offset = (const float*)d_in[1];   // (b, 3, c)
    const float* conv_w = (const float*)d_in[2];   // (3,)
    float*       out    = (float*)d_out;           // (b, 1, c)

    const int b = in_sizes[0] / C_DIM;             // 2048 for the reference setup
    deform_conv1d_kernel<<<dim3(b), dim3(BLOCK), 0, stream>>>(x, offset, conv_w, out);
}